// Gate_52243982188858
// MI455X (gfx1250) — compile-verified
//
#include <hip/hip_runtime.h>
#include <math.h>

// MoE top-2 gating for MI455X (gfx1250, wave32).
// x: [16384, 8192] fp32, W: [8, 8192] fp32 -> out: [16384, 8] fp32 dense top-2 softmax weights.
// HBM-bound (512MB x stream, ~22us floor at 23.3 TB/s). Matrix math via V_WMMA_F32_16X16X4_F32.
// Split-K across 8 waves/block -> 4096 waves total for latency coverage (~4 waves/SIMD).

typedef float v2f __attribute__((ext_vector_type(2)));
typedef float v8f __attribute__((ext_vector_type(8)));

#define D_DIM        8192
#define E_DIM        8
#define TOK_PER_BLK  32                    // two 16x16 WMMA C-tiles, shared B operand
#define WAVES_PER_BLK 8
#define K_SLICE      (D_DIM / WAVES_PER_BLK)   // 1024 columns per wave

__global__ __launch_bounds__(256)
void moe_gate_wmma_kernel(const float* __restrict__ x,
                          const float* __restrict__ W,
                          float* __restrict__ out) {
    // Per-wave partial logits: [wave][token-in-block 0..31][expert 0..7] = 8 KB.
    __shared__ float partial[WAVES_PER_BLK * TOK_PER_BLK * E_DIM];

    const int tid    = threadIdx.x;
    const int wave   = tid >> 5;             // 0..7 -> K-slice owner
    const int lane   = tid & 31;             // full wave: EXEC all ones for WMMA
    const int mlo    = lane & 15;            // A: row index / B,C: column index
    const int khalf  = lane >> 4;            // 0: K=0,1  1: K=2,3 (32-bit WMMA layout)
    const int token0 = blockIdx.x * TOK_PER_BLK;
    const int kbase  = wave * K_SLICE;

    // A-matrix per-lane stream: lane l and l+16 both read token row (l&15),
    // at K-offsets {0,1}/{2,3} of the current 4-wide K chunk.
    const float* aPtr0 = x + (size_t)(token0 + mlo) * D_DIM + kbase + 2 * khalf;
    const float* aPtr1 = aPtr0 + (size_t)16 * D_DIM;   // second 16-token tile

    // B-matrix: column n reads expert row n. Columns 8..15 mirror experts 0..7
    // (same addresses -> coalesced, no masking in the hot loop); duplicate
    // output columns are ignored in the epilogue.
    const float* bPtr = W + (size_t)(mlo & 7) * D_DIM + kbase + 2 * khalf;

    v8f acc0 = {0.f, 0.f, 0.f, 0.f, 0.f, 0.f, 0.f, 0.f};
    v8f acc1 = {0.f, 0.f, 0.f, 0.f, 0.f, 0.f, 0.f, 0.f};

    // K-slice loop: 1024/4 = 256 WMMA steps, unrolled x8 so the compiler
    // batches global_load_b64s (imm offsets, s_clause) ahead of the WMMA chain.
#pragma unroll 8
    for (int d = 0; d < K_SLICE; d += 4) {
        v2f a0 = *(const v2f*)(aPtr0 + d);   // global_load_b64 (A tile 0, HBM stream)
        v2f a1 = *(const v2f*)(aPtr1 + d);   // global_load_b64 (A tile 1, HBM stream)
        v2f b  = *(const v2f*)(bPtr  + d);   // global_load_b64 (B, L2-hot)
        acc0 = __builtin_amdgcn_wmma_f32_16x16x4_f32(
            false, a0, false, b, (short)0, acc0, false, false);
        acc1 = __builtin_amdgcn_wmma_f32_16x16x4_f32(
            false, a1, false, b, (short)0, acc1, false, false);
    }

    // ---- Stage per-wave partial C tiles to LDS ----
    // C/D layout (32-bit, 16x16): VGPR r -> lanes 0..15: M=r, N=lane;
    //                                       lanes 16..31: M=8+r, N=lane-16.
    const int e = mlo;                       // expert column this lane holds
    float* myPart = partial + wave * (TOK_PER_BLK * E_DIM);
#pragma unroll
    for (int r = 0; r < 8; ++r) {
        const int tl = (lane < 16) ? r : (r + 8);   // token-in-block
        if (e < E_DIM) {                            // skip duplicate columns 8..15
            myPart[tl * E_DIM + e]        = acc0[r];
            myPart[(tl + 16) * E_DIM + e] = acc1[r];
        }
    }
    __syncthreads();

    // ---- Deterministic split-K reduction: thread tid owns (token=tid/8, e=tid%8) ----
    float s = 0.f;
#pragma unroll
    for (int w = 0; w < WAVES_PER_BLK; ++w)
        s += partial[w * (TOK_PER_BLK * E_DIM) + tid];
    partial[tid] = s;   // no other thread reads slot tid before the barrier (w=0 term already consumed)
    __syncthreads();

    // ---- Top-2 softmax + dense scatter: one token per lane of wave 0 ----
    if (tid < TOK_PER_BLK) {
        float l[E_DIM];
#pragma unroll
        for (int ei = 0; ei < E_DIM; ++ei) l[ei] = partial[tid * E_DIM + ei];

        int i1 = 0; float m1 = l[0];
#pragma unroll
        for (int ei = 1; ei < E_DIM; ++ei)
            if (l[ei] > m1) { m1 = l[ei]; i1 = ei; }   // strict '>' => lowest index on ties

        int i2 = 0; float m2 = -INFINITY;
#pragma unroll
        for (int ei = 0; ei < E_DIM; ++ei)
            if (ei != i1 && l[ei] > m2) { m2 = l[ei]; i2 = ei; }

        const float e2 = __expf(m2 - m1);    // m1 >= m2, numerically safe
        const float ss = 1.0f + e2;
        const float w1 = 1.0f / ss;
        const float w2 = e2 / ss;

        float o[E_DIM];
#pragma unroll
        for (int ei = 0; ei < E_DIM; ++ei)
            o[ei] = (ei == i1) ? w1 : ((ei == i2) ? w2 : 0.0f);

        float4* op = (float4*)(out + (size_t)(token0 + tid) * E_DIM);
        op[0] = make_float4(o[0], o[1], o[2], o[3]);
        op[1] = make_float4(o[4], o[5], o[6], o[7]);
    }
}

extern "C" void kernel_launch(void* const* d_in, const int* in_sizes, int n_in,
                              void* d_out, int out_size, void* d_ws, size_t ws_size,
                              hipStream_t stream) {
    const float* x = (const float*)d_in[0];   // [N, 8192]
    const float* W = (const float*)d_in[1];   // [8, 8192]
    float* out = (float*)d_out;               // [N, 8]

    const int n_tokens = in_sizes[0] / D_DIM;           // 16384
    const int n_blocks = n_tokens / TOK_PER_BLK;        // 512 blocks x 8 waves = 4096 waves

    hipLaunchKernelGGL(moe_gate_wmma_kernel, dim3(n_blocks), dim3(256), 0, stream,
                       x, W, out);
}